// MegaBlocksMoeBlock_40733469835723
// MI455X (gfx1250) — compile-verified
//
#include <hip/hip_runtime.h>
#include <hip/hip_bf16.h>
#include <math.h>

// ---------------------------------------------------------------------------
// MoE forward for MI455X (gfx1250, wave32, WMMA f32 16x16x4).
// D=F=1024, E=8, K=2, N=B*T=4096. Double-buffered LDS staging via CDNA5
// GLOBAL_LOAD_ASYNC_TO_LDS (ASYNCcnt). Block tile 128x128, wave tile 32x64.
// ---------------------------------------------------------------------------

typedef __attribute__((ext_vector_type(2))) float v2f;
typedef __attribute__((ext_vector_type(8))) float v8f;
typedef __attribute__((ext_vector_type(4))) int   v4i;

#define MOE_D 1024
#define MOE_F 1024
#define MOE_E 8
#define MOE_N 4096

// ---- CDNA5 async global->LDS path (probe via __has_builtin) ---------------
#if defined(__has_builtin)
#if __has_builtin(__builtin_amdgcn_global_load_async_to_lds_b128) && \
    __has_builtin(__builtin_amdgcn_s_wait_asynccnt)
#define USE_ASYNC_LDS 1
#endif
#endif
#ifndef USE_ASYNC_LDS
#define USE_ASYNC_LDS 0
#endif

#if USE_ASYNC_LDS
typedef __attribute__((address_space(1))) v4i glb_v4i;
typedef __attribute__((address_space(3))) v4i lds_v4i;
__device__ __forceinline__ void stage16(const float* __restrict__ g, float* l) {
    // per-lane: copies 16 bytes from global address g to LDS address l, tracked by ASYNCcnt
    __builtin_amdgcn_global_load_async_to_lds_b128(
        (glb_v4i*)g, (lds_v4i*)l, 0, 0);
}
#else
__device__ __forceinline__ void stage16(const float* __restrict__ g, float* l) {
    *reinterpret_cast<float4*>(l) = *reinterpret_cast<const float4*>(g);
}
#endif

__device__ __forceinline__ float silu_f(float v) {
    return v / (1.0f + __expf(-v));
}

// ---------------------------------------------------------------------------
// Zero-fill
// ---------------------------------------------------------------------------
__global__ void zero_kernel(float* __restrict__ p, int n) {
    int i = blockIdx.x * blockDim.x + threadIdx.x;
    if (i < n) p[i] = 0.0f;
}

// ---------------------------------------------------------------------------
// Router: one wave32 per token. logits = x@router_w (D x 8), softmax, top-2
// scattered into combine[n][8]; also sgate[n] = sigmoid(x @ shared_gate_w).
// ---------------------------------------------------------------------------
__global__ __launch_bounds__(256)
void router_kernel(const float* __restrict__ x,
                   const float* __restrict__ rw,     // [D][E]
                   const float* __restrict__ sgw,    // [D]
                   float* __restrict__ combine,      // [N][E]
                   float* __restrict__ sgate,        // [N]
                   int N, int D) {
    const int wave = threadIdx.x >> 5;
    const int lane = threadIdx.x & 31;
    const int n = blockIdx.x * 8 + wave;
    if (n >= N) return;

    const float* xr = x + (size_t)n * D;
    float acc[MOE_E];
#pragma unroll
    for (int e = 0; e < MOE_E; ++e) acc[e] = 0.0f;
    float accg = 0.0f;

    for (int d = lane; d < D; d += 32) {
        float xv = xr[d];
#pragma unroll
        for (int e = 0; e < MOE_E; ++e) acc[e] = fmaf(xv, rw[d * MOE_E + e], acc[e]);
        accg = fmaf(xv, sgw[d], accg);
    }
#pragma unroll
    for (int off = 16; off > 0; off >>= 1) {
#pragma unroll
        for (int e = 0; e < MOE_E; ++e) acc[e] += __shfl_xor(acc[e], off, 32);
        accg += __shfl_xor(accg, off, 32);
    }
    if (lane == 0) {
        float mx = acc[0];
#pragma unroll
        for (int e = 1; e < MOE_E; ++e) mx = fmaxf(mx, acc[e]);
        float p[MOE_E];
        float sum = 0.0f;
#pragma unroll
        for (int e = 0; e < MOE_E; ++e) { p[e] = __expf(acc[e] - mx); sum += p[e]; }
        float inv = 1.0f / sum;
#pragma unroll
        for (int e = 0; e < MOE_E; ++e) p[e] *= inv;
        int i1 = 0;
#pragma unroll
        for (int e = 1; e < MOE_E; ++e) if (p[e] > p[i1]) i1 = e;
        int i2 = (i1 == 0) ? 1 : 0;
#pragma unroll
        for (int e = 0; e < MOE_E; ++e) if (e != i1 && p[e] > p[i2]) i2 = e;
        float* cr = combine + (size_t)n * MOE_E;
#pragma unroll
        for (int e = 0; e < MOE_E; ++e)
            cr[e] = (e == i1) ? p[i1] : ((e == i2) ? p[i2] : 0.0f);
        sgate[n] = 1.0f / (1.0f + __expf(-accg));
    }
}

// ---------------------------------------------------------------------------
// Tiled f32 WMMA GEMM, C-tile 128x128 per 256-thread block (8 waves, wave32,
// wave grid 4x2). Each wave owns 32x64 = 2x4 WMMA 16x16 tiles, K-step 4 per
// WMMA: 8 WMMAs per A-fragment pair. Double-buffered LDS; staging via async
// global->LDS when available.
// ---------------------------------------------------------------------------
#define MODE_SILU_SCALE 0
#define MODE_ADD        1
#define MODE_GATED      2
#define MODE_SCALE_ADD  3

template <int MODE>
__global__ __launch_bounds__(256)
void gemm_wmma_f32(const float* __restrict__ A, int lda,
                   const float* __restrict__ B, int ldb, int upOff,
                   float* __restrict__ C, int ldc,
                   const float* __restrict__ scale, int scaleStride,
                   int M, int N, int K) {
    constexpr int BM = 128, BN = 128, BK = 16;
    constexpr int NG = (MODE == MODE_GATED) ? 2 : 1;   // gate/up streams
    constexpr int ASTRIDE = BK + 4;    // 20 words: conflict-free A frag reads
    constexpr int BSTRIDE = BN + 16;   // 144 words: 2*144 % 64 == 32 -> clean
    // async ops issued per wave per tile (in-order completion on ASYNCcnt)
    constexpr int PEND = 2 + 2 * NG;

    __shared__ float As[2][BM][ASTRIDE];
    __shared__ float Bs[2][NG][BK][BSTRIDE];

    const int tid  = threadIdx.x;
    const int wave = tid >> 5;
    const int lane = tid & 31;
    const int lo   = lane & 15;
    const int hi   = lane >> 4;

    const int wm0 = (wave >> 1) << 5;  // 0,32,64,96
    const int wn0 = (wave & 1) << 6;   // 0,64

    const int m0 = blockIdx.y * BM;
    const int n0 = blockIdx.x * BN;

    // staging coordinates (fixed per thread)
    const int ar = tid >> 2;            // A row   0..63 (+64 second half)
    const int ac = (tid & 3) << 2;      // A col   0,4,8,12
    const int br = tid >> 5;            // B row   0..7 (+8 on second half)
    const int bc = (tid & 31) << 2;     // B col   0..124

    v8f acc[NG][2][4];
#pragma unroll
    for (int g = 0; g < NG; ++g)
#pragma unroll
        for (int i = 0; i < 2; ++i)
#pragma unroll
            for (int j = 0; j < 4; ++j)
                acc[g][i][j] = (v8f)(0.0f);

    // issue staging for tile at k0 into buffer `buf`
    auto stage_tile = [&](int buf, int k0) {
        stage16(&A[(size_t)(m0 + ar) * lda + k0 + ac], &As[buf][ar][ac]);
        stage16(&A[(size_t)(m0 + ar + 64) * lda + k0 + ac], &As[buf][ar + 64][ac]);
#pragma unroll
        for (int g = 0; g < NG; ++g)
#pragma unroll
            for (int h = 0; h < 2; ++h)
                stage16(&B[(size_t)(k0 + br + h * 8) * ldb + n0 + g * upOff + bc],
                        &Bs[buf][g][br + h * 8][bc]);
    };

    stage_tile(0, 0);
    int cur = 0;
    for (int k0 = 0; k0 < K; k0 += BK) {
        const bool has_next = (k0 + BK) < K;
        if (has_next) stage_tile(cur ^ 1, k0 + BK);
#if USE_ASYNC_LDS
        // tile `cur` complete; next tile's PEND ops may remain in flight
        if (has_next) __builtin_amdgcn_s_wait_asynccnt(PEND);
        else          __builtin_amdgcn_s_wait_asynccnt(0);
#endif
        __syncthreads();

#pragma unroll
        for (int kk = 0; kk < BK; kk += 4) {
            const int kb = kk + (hi << 1);
            const v2f a0 = *reinterpret_cast<const v2f*>(&As[cur][wm0 + lo][kb]);
            const v2f a1 = *reinterpret_cast<const v2f*>(&As[cur][wm0 + 16 + lo][kb]);
#pragma unroll
            for (int g = 0; g < NG; ++g) {
#pragma unroll
                for (int j = 0; j < 4; ++j) {
                    const int nn = wn0 + j * 16 + lo;
                    v2f b;
                    b.x = Bs[cur][g][kb][nn];
                    b.y = Bs[cur][g][kb + 1][nn];
                    acc[g][0][j] = __builtin_amdgcn_wmma_f32_16x16x4_f32(
                        false, a0, false, b, (short)0, acc[g][0][j], false, false);
                    acc[g][1][j] = __builtin_amdgcn_wmma_f32_16x16x4_f32(
                        false, a1, false, b, (short)0, acc[g][1][j], false, false);
                }
            }
        }
        __syncthreads();   // protects buf cur^1 (next issue target)
        cur ^= 1;
    }

    // --- epilogue: lane l, vgpr v -> row = v + 8*hi, col = lo ---
#pragma unroll
    for (int i = 0; i < 2; ++i) {
#pragma unroll
        for (int j = 0; j < 4; ++j) {
            union { v8f v; float f[8]; } u0, u1;
            u0.v = acc[0][i][j];
            if constexpr (MODE == MODE_GATED) u1.v = acc[NG - 1][i][j];
#pragma unroll
            for (int v = 0; v < 8; ++v) {
                const int grow = m0 + wm0 + i * 16 + v + 8 * hi;
                const int gcol = n0 + wn0 + j * 16 + lo;
                float* cp = &C[(size_t)grow * ldc + gcol];
                if constexpr (MODE == MODE_SILU_SCALE) {
                    const float s = scale[(size_t)grow * scaleStride];
                    *cp = silu_f(u0.f[v]) * s;
                } else if constexpr (MODE == MODE_ADD) {
                    *cp += u0.f[v];
                } else if constexpr (MODE == MODE_GATED) {
                    *cp = silu_f(u0.f[v]) * u1.f[v];
                } else { // MODE_SCALE_ADD
                    const float s = scale[(size_t)grow * scaleStride];
                    *cp = fmaf(s, u0.f[v], *cp);
                }
            }
        }
    }
}

// ---------------------------------------------------------------------------
// Host launcher
// ---------------------------------------------------------------------------
extern "C" void kernel_launch(void* const* d_in, const int* in_sizes, int n_in,
                              void* d_out, int out_size, void* d_ws, size_t ws_size,
                              hipStream_t stream) {
    const float* x    = (const float*)d_in[0]; // [N][D]
    const float* rw   = (const float*)d_in[1]; // [D][E]
    const float* w1   = (const float*)d_in[2]; // [E][D][F]
    const float* w2   = (const float*)d_in[3]; // [E][F][D]
    const float* sgu  = (const float*)d_in[4]; // [D][2F]
    const float* sdn  = (const float*)d_in[5]; // [F][D]
    const float* sgw  = (const float*)d_in[6]; // [D][1]
    float* out = (float*)d_out;                // [N][D]

    const int D = MOE_D, F = MOE_F, E = MOE_E, N = MOE_N;

    float* ws      = (float*)d_ws;
    float* H       = ws;                        // [N][F] scratch activations
    float* combine = H + (size_t)N * F;         // [N][E]
    float* sgate   = combine + (size_t)N * E;   // [N]

    zero_kernel<<<(N * D + 255) / 256, 256, 0, stream>>>(out, N * D);
    router_kernel<<<N / 8, 256, 0, stream>>>(x, rw, sgw, combine, sgate, N, D);

    const dim3 blk(256);
    const dim3 gridF(F / 128, N / 128);
    const dim3 gridD(D / 128, N / 128);

    // shared expert: t = silu(x@Wg) * (x@Wu);  out += sgate * (t @ Wdown)
    gemm_wmma_f32<MODE_GATED><<<gridF, blk, 0, stream>>>(
        x, D, sgu, 2 * F, F, H, F, nullptr, 0, N, F, D);
    gemm_wmma_f32<MODE_SCALE_ADD><<<gridD, blk, 0, stream>>>(
        H, F, sdn, D, 0, out, D, sgate, 1, N, D, F);

    // routed experts (combine weight folded into H; zero rows contribute 0)
    for (int e = 0; e < E; ++e) {
        gemm_wmma_f32<MODE_SILU_SCALE><<<gridF, blk, 0, stream>>>(
            x, D, w1 + (size_t)e * D * F, F, 0, H, F, combine + e, E, N, F, D);
        gemm_wmma_f32<MODE_ADD><<<gridD, blk, 0, stream>>>(
            H, F, w2 + (size_t)e * F * D, D, 0, out, D, nullptr, 0, N, D, F);
    }
}